// ScaledDotProductAttention_37031208026419
// MI455X (gfx1250) — compile-verified
//
#include <hip/hip_runtime.h>

// Problem constants (from reference): B=8, H=8, S=16384, Dv=64, d_k=1
#define NHEADS 64      // B*H
#define SEQ    16384
#define DV     64
#define SPLIT  32              // S-chunks per head in the reduction pass
#define ROWS   (SEQ / SPLIT)   // 512 rows per chunk

typedef __attribute__((ext_vector_type(2))) float v2f;
typedef __attribute__((ext_vector_type(8))) float v8f;

// ---------------------------------------------------------------------------
// Pass 1: per head, partial kv[n] = sum_s K[s]*V[s,n], plus sum(K^2), sum(Q^2).
// Grid (NHEADS, SPLIT), block 256. Thread t owns column t%64, rows strided by 4.
// V is streamed fully coalesced (256 consecutive floats per 8 waves per iter).
// ---------------------------------------------------------------------------
__global__ void reduce_kernel(const float* __restrict__ Q,
                              const float* __restrict__ K,
                              const float* __restrict__ V,
                              float* __restrict__ kvacc,   // [NHEADS*64]
                              float* __restrict__ k2acc,   // [NHEADS]
                              float* __restrict__ q2acc) { // [NHEADS]
  const int bh    = blockIdx.x;
  const int chunk = blockIdx.y;
  const int t     = threadIdx.x;           // 0..255
  const int col   = t & 63;
  const int rsub  = t >> 6;                // 0..3
  const int sbase = chunk * ROWS;

  const float* __restrict__ Kh = K + (size_t)bh * SEQ;
  const float* __restrict__ Qh = Q + (size_t)bh * SEQ;
  const float* __restrict__ Vh = V + (size_t)bh * SEQ * DV;

  float acc = 0.0f;
  for (int r = rsub; r < ROWS; r += 4) {
    const int s = sbase + r;
    // prefetch the V stream ~32 rows ahead (emits global_prefetch_b8)
    int rp = r + 32; if (rp > ROWS - 1) rp = ROWS - 1;
    __builtin_prefetch(&Vh[(size_t)(sbase + rp) * DV + col], 0, 1);
    acc = fmaf(Kh[s], Vh[(size_t)s * DV + col], acc);
  }

  // norm partials over disjoint rows per thread
  float k2 = 0.0f, q2 = 0.0f;
  for (int r = t; r < ROWS; r += 256) {
    const float kv = Kh[sbase + r];
    const float qv = Qh[sbase + r];
    k2 = fmaf(kv, kv, k2);
    q2 = fmaf(qv, qv, q2);
  }

  __shared__ float sacc[256];
  __shared__ float sk2[256];
  __shared__ float sq2[256];
  sacc[t] = acc; sk2[t] = k2; sq2[t] = q2;
  __syncthreads();

  if (t < 64) {
    const float v = sacc[t] + sacc[t + 64] + sacc[t + 128] + sacc[t + 192];
    atomicAdd(&kvacc[bh * 64 + t], v);
  }
  for (int off = 128; off > 0; off >>= 1) {
    __syncthreads();
    if (t < off) { sk2[t] += sk2[t + off]; sq2[t] += sq2[t + off]; }
  }
  if (t == 0) {
    atomicAdd(&k2acc[bh], sk2[0]);
    atomicAdd(&q2acc[bh], sq2[0]);
  }
}

// ---------------------------------------------------------------------------
// Pass 1b: fold norms into kv and produce 1/||Q||. Grid (NHEADS), block 64.
// (1/sqrt(d_k) == 1 for d_k=1, so it is omitted.)
// ---------------------------------------------------------------------------
__global__ void finalize_kernel(const float* __restrict__ kvacc,
                                const float* __restrict__ k2acc,
                                const float* __restrict__ q2acc,
                                float* __restrict__ kvs,    // [NHEADS*64]
                                float* __restrict__ qinv) { // [NHEADS]
  const int bh = blockIdx.x;
  const int n  = threadIdx.x;  // 0..63
  const float kinv = rsqrtf(k2acc[bh]);
  kvs[bh * 64 + n] = kvacc[bh * 64 + n] * kinv;
  if (n == 0) qinv[bh] = rsqrtf(q2acc[bh]);
}

// ---------------------------------------------------------------------------
// Pass 2: out[s, n] = (Q[s] * qinv) * kvs[n] as a rank-1 WMMA:
//   D(16x16) = A(16x4) x B(4x16),  A col 0 = scaled Q rows, B row 0 = kv tile,
//   all other K slots zero.
// A layout (f32 16x4): VGPR0 lanes 0-15 = K=0 (M=lane); lanes 16-31 = K=2;
//                      VGPR1 = K=1 / K=3  -> zero everywhere except a.x lane<16.
// B layout: row K=0 lives in VGPR0 lanes 0-15 (N=lane); all else zero.
// D layout: VGPR i -> (M=i, N=lane) for lanes 0-15, (M=i+8, N=lane-16) for 16-31.
// Grid (NHEADS, SEQ/128), block 256 (8 waves x 16 rows each), 4 N-tiles of 16.
// ---------------------------------------------------------------------------
__global__ void outer_kernel(const float* __restrict__ Q,
                             const float* __restrict__ kvs,
                             const float* __restrict__ qinv,
                             float* __restrict__ out) {
  const int bh   = blockIdx.x;
  const int wave = threadIdx.x >> 5;   // 0..7
  const int lane = threadIdx.x & 31;
  const int s0   = blockIdx.y * 128 + wave * 16;

  const float qscale = qinv[bh];
  const float* __restrict__ ktv  = kvs + bh * 64;
  float* __restrict__       outh = out + (size_t)bh * SEQ * DV;

  v2f a; a.x = 0.0f; a.y = 0.0f;
  if (lane < 16) a.x = Q[(size_t)bh * SEQ + s0 + lane] * qscale;

  const int Mbase = (lane >= 16) ? 8 : 0;
  const int N     = lane & 15;

#pragma unroll
  for (int tile = 0; tile < 4; ++tile) {
    v2f b; b.x = 0.0f; b.y = 0.0f;
    if (lane < 16) b.x = ktv[tile * 16 + lane];
    v8f c = {};
    c = __builtin_amdgcn_wmma_f32_16x16x4_f32(
        /*neg_a=*/false, a, /*neg_b=*/false, b,
        /*c_mod=*/(short)0, c, /*reuse_a=*/false, /*reuse_b=*/false);
#pragma unroll
    for (int i = 0; i < 8; ++i) {
      const int m = s0 + Mbase + i;
      outh[(size_t)m * DV + tile * 16 + N] = c[i];
    }
  }
}

// ---------------------------------------------------------------------------
// Workspace layout (floats):
//   [0      , 4096)  kvacc   (accumulated, must be zeroed)
//   [4096   , 4160)  k2acc   (zeroed)
//   [4160   , 4224)  q2acc   (zeroed)
//   [4224   , 8320)  kvs
//   [8320   , 8384)  qinv
// ---------------------------------------------------------------------------
extern "C" void kernel_launch(void* const* d_in, const int* in_sizes, int n_in,
                              void* d_out, int out_size, void* d_ws, size_t ws_size,
                              hipStream_t stream) {
  (void)in_sizes; (void)n_in; (void)out_size; (void)ws_size;
  const float* Q = (const float*)d_in[0];
  const float* K = (const float*)d_in[1];
  const float* V = (const float*)d_in[2];
  float* out = (float*)d_out;

  float* ws    = (float*)d_ws;
  float* kvacc = ws;
  float* k2acc = ws + 4096;
  float* q2acc = ws + 4160;
  float* kvs   = ws + 4224;
  float* qinv  = ws + 8320;

  // zero the accumulators every call (harness does not re-poison between replays)
  hipMemsetAsync(kvacc, 0, (size_t)4224 * sizeof(float), stream);

  reduce_kernel<<<dim3(NHEADS, SPLIT), 256, 0, stream>>>(Q, K, V, kvacc, k2acc, q2acc);
  finalize_kernel<<<dim3(NHEADS), 64, 0, stream>>>(kvacc, k2acc, q2acc, kvs, qinv);
  outer_kernel<<<dim3(NHEADS, SEQ / 128), 256, 0, stream>>>(Q, kvs, qinv, out);
}